// DotPredictor_5411658793098
// MI455X (gfx1250) — compile-verified
//
#include <hip/hip_runtime.h>
#include <cstdint>

#define WAVES_PER_BLOCK 8
#define EDGES_PER_WAVE  8
#define NBUF            4      // async pipeline depth (edges in flight)
#define DFEAT           128

// ---------------------------------------------------------------------------
// Full-tile kernel: every wave processes exactly EDGES_PER_WAVE edges via a
// 4-deep async global->LDS pipeline. 32-bit saddr-form addressing (h table is
// 25.6MB, row offsets fit 32 bits), deferred multi-value tree reduction.
// ---------------------------------------------------------------------------
__global__ __launch_bounds__(WAVES_PER_BLOCK * 32)
void edge_dot_async_full(const float* __restrict__ h,
                         const int*  __restrict__ src,
                         const int*  __restrict__ dst,
                         float*      __restrict__ out)
{
    // [wave][buffer][u/v row][128 floats] -> 32 KB per 256-thread block
    __shared__ float lds[WAVES_PER_BLOCK][NBUF][2][DFEAT];

    const int lane  = threadIdx.x & 31;
    const int wline = threadIdx.x >> 5;
    const int e0    = (blockIdx.x * WAVES_PER_BLOCK + wline) * EDGES_PER_WAVE;

    // Preload all edge indices into registers (vectorizes to b128 clauses).
    int rs[EDGES_PER_WAVE], rd[EDGES_PER_WAVE];
    #pragma unroll
    for (int i = 0; i < EDGES_PER_WAVE; ++i) {
        rs[i] = src[e0 + i];
        rd[i] = dst[e0 + i];
    }

    const uint64_t hbase   = (uint64_t)(uintptr_t)h;
    const uint32_t laneoff = (uint32_t)lane * 16u;

    // Issue async loads of both 512B rows for local edge i into LDS buffer b.
    // saddr form: 64-bit SGPR base + 32-bit per-lane byte offset (row*512+lane*16).
    auto issue = [&](int i, int b) {
        uint32_t offu = ((uint32_t)rs[i] << 9) + laneoff;
        uint32_t offv = ((uint32_t)rd[i] << 9) + laneoff;
        // Flat shared-pointer low 32 bits == group-segment byte offset (ISA 10.2).
        uint32_t lu = (uint32_t)(uintptr_t)(&lds[wline][b][0][lane * 4]);
        uint32_t lv = (uint32_t)(uintptr_t)(&lds[wline][b][1][lane * 4]);
        asm volatile("global_load_async_to_lds_b128 %0, %1, %2"
                     :: "v"(lu), "v"(offu), "s"(hbase) : "memory");
        asm volatile("global_load_async_to_lds_b128 %0, %1, %2"
                     :: "v"(lv), "v"(offv), "s"(hbase) : "memory");
    };

    // Prologue: fill NBUF-1 pipeline stages.
    #pragma unroll
    for (int i = 0; i < NBUF - 1; ++i)
        issue(i, i);

    float s[EDGES_PER_WAVE];

    #pragma unroll
    for (int i = 0; i < EDGES_PER_WAVE; ++i) {
        const int b = i & (NBUF - 1);

        // Keep the pipeline full (buffer re-targeted here was consumed at i-1).
        if (i + (NBUF - 1) < EDGES_PER_WAVE)
            issue(i + (NBUF - 1), (i + (NBUF - 1)) & (NBUF - 1));

        // Async completes in order: <= 2*ahead outstanding ==> edge i's loads done.
        // i is an unroll constant, so exactly one literal wait survives per stage.
        const int ahead = (EDGES_PER_WAVE - 1 - i) < (NBUF - 1)
                              ? (EDGES_PER_WAVE - 1 - i) : (NBUF - 1);
        switch (ahead) {
            case 3:  asm volatile("s_wait_asynccnt 6" ::: "memory"); break;
            case 2:  asm volatile("s_wait_asynccnt 4" ::: "memory"); break;
            case 1:  asm volatile("s_wait_asynccnt 2" ::: "memory"); break;
            default: asm volatile("s_wait_asynccnt 0" ::: "memory"); break;
        }

        const float4 u = *(const float4*)(&lds[wline][b][0][lane * 4]);
        const float4 v = *(const float4*)(&lds[wline][b][1][lane * 4]);
        s[i] = u.x * v.x + u.y * v.y + u.z * v.z + u.w * v.w;
    }

    // Multi-value wave32 tree reduction of 8 per-edge partial sums:
    // 16 shuffles + 16 adds + 7 selects (vs 40+40 for 8 separate butterflies).
    // Packing encodes edge index in lane bits: bit4->edge b0, bit3->b1, bit2->b2.
    float t[8];
    #pragma unroll
    for (int i = 0; i < 8; ++i)
        t[i] = s[i] + __shfl_xor(s[i], 16, 32);     // replicated across bit4

    float p[4];
    #pragma unroll
    for (int j = 0; j < 4; ++j) {
        p[j] = (lane & 16) ? t[2 * j + 1] : t[2 * j];
        p[j] += __shfl_xor(p[j], 8, 32);            // replicated across bit3
    }

    float r[2];
    #pragma unroll
    for (int k = 0; k < 2; ++k) {
        r[k] = (lane & 8) ? p[2 * k + 1] : p[2 * k];
        r[k] += __shfl_xor(r[k], 4, 32);            // replicated across bit2
    }

    float w = (lane & 4) ? r[1] : r[0];
    w += __shfl_xor(w, 2, 32);
    w += __shfl_xor(w, 1, 32);                      // w = total for edge(lane)

    if ((lane & 3) == 0) {
        const int edge = ((lane >> 4) & 1) | (((lane >> 3) & 1) << 1)
                       | (((lane >> 2) & 1) << 2);
        out[e0 + edge] = w;
    }
}

// ---------------------------------------------------------------------------
// Tail kernel: wave-per-edge, direct global b128 loads (< 64 edges).
// ---------------------------------------------------------------------------
__global__ __launch_bounds__(256)
void edge_dot_tail(const float* __restrict__ h,
                   const int*  __restrict__ src,
                   const int*  __restrict__ dst,
                   float*      __restrict__ out,
                   int e_base, int n_edges)
{
    const int lane = threadIdx.x & 31;
    const int e    = e_base + blockIdx.x * (blockDim.x >> 5) + (threadIdx.x >> 5);
    if (e >= n_edges) return;

    const float4 u = *(const float4*)(h + (size_t)src[e] * DFEAT + lane * 4);
    const float4 v = *(const float4*)(h + (size_t)dst[e] * DFEAT + lane * 4);
    float s = u.x * v.x + u.y * v.y + u.z * v.z + u.w * v.w;

    #pragma unroll
    for (int off = 16; off > 0; off >>= 1)
        s += __shfl_xor(s, off, 32);

    if (lane == 0) out[e] = s;
}

extern "C" void kernel_launch(void* const* d_in, const int* in_sizes, int n_in,
                              void* d_out, int out_size, void* d_ws, size_t ws_size,
                              hipStream_t stream) {
    (void)n_in; (void)out_size; (void)d_ws; (void)ws_size;
    const float* h   = (const float*)d_in[0];
    const int*   src = (const int*)d_in[1];
    const int*   dst = (const int*)d_in[2];
    float*       out = (float*)d_out;

    const int n_edges       = in_sizes[1];
    const int edges_per_blk = WAVES_PER_BLOCK * EDGES_PER_WAVE;   // 64
    const int full_blocks   = n_edges / edges_per_blk;
    const int covered       = full_blocks * edges_per_blk;
    const int tail          = n_edges - covered;

    if (full_blocks > 0)
        edge_dot_async_full<<<full_blocks, WAVES_PER_BLOCK * 32, 0, stream>>>(
            h, src, dst, out);

    if (tail > 0) {
        const int tail_blocks = (tail + 7) / 8;  // 8 waves (edges) per 256-thr block
        edge_dot_tail<<<tail_blocks, 256, 0, stream>>>(
            h, src, dst, out, covered, n_edges);
    }
}